// QuantumLayer_9483287790225
// MI455X (gfx1250) — compile-verified
//
#include <hip/hip_runtime.h>
#include <math.h>

typedef __attribute__((ext_vector_type(2))) float v2f;
typedef __attribute__((ext_vector_type(8))) float v8f;

#define NQ 14
#define PI_F 3.14159265358979323846f

// One wave (32 lanes) per 16-batch-row tile.
//
// Math: the circuit's single-qubit gates act on a product state; the CNOT ring
// is a GF(2) basis permutation absorbed into the <Z_i> readout, giving
//   z(b,j) = cos(a)*(cos(w) - sin(w)*sin(a)),  a = pi*x[b,j], w = weight[j]
//   out[b][i] = prod_{j in S_i} z(b,j),  S_i = {0..i} (i>=1), S_0 = {1..13}.
// The masked products are evaluated as two f32 WMMA matmuls in log domain
// (log-magnitude sum + negative-sign count) against a constant 16x16 0/1 mask.
__global__ __launch_bounds__(32)
void quantum_layer_wmma_kernel(const float* __restrict__ x,
                               const float* __restrict__ weight,
                               float* __restrict__ out,
                               int batch) {
  const int lane = threadIdx.x & 31;
  const int m    = lane & 15;   // batch row within tile (A-M) / output col (B-N)
  const int hi   = lane >> 4;   // lane half selects K pairs per ISA layout
  const int rowBase = blockIdx.x * 16;
  const int b = rowBase + m;
  const int row = (b < batch) ? b : (batch - 1);  // clamp: rows beyond batch are
                                                  // independent M rows, never stored

  // Mask^T row bitmask for this lane's N column (bit k set <=> qubit k in S_n).
  const unsigned mbits = (m == 0) ? 0x3FFEu
                       : ((m < NQ) ? ((2u << m) - 1u) : 0u);

  // Build WMMA operands in registers. A 16x4 f32 layout (ISA 7.12.2):
  // VGPR0 holds K = 2*hi, VGPR1 holds K = 1+2*hi; chunk kc adds 4*kc.
  v2f AL[4];  // log|z|  (M=batch row, K=qubit; K>=14 padded 0)
  v2f AF[4];  // 1.0 where z<0
  v2f BM[4];  // Mask^T  (K=qubit j, N=output i)
#pragma unroll
  for (int kc = 0; kc < 4; ++kc) {
#pragma unroll
    for (int t = 0; t < 2; ++t) {
      const int k = 4 * kc + t + 2 * hi;          // K index in this VGPR slot
      const int j = (k < NQ) ? k : (NQ - 1);      // clamped, in-bounds load
      const float a = PI_F * x[row * NQ + j];
      const float w = weight[j];
      const float z = __cosf(a) * (__cosf(w) - __sinf(w) * __sinf(a));
      const float Lv = __logf(fmaxf(fabsf(z), 1e-30f));
      const bool valid = (k < NQ);
      AL[kc][t] = valid ? Lv : 0.0f;
      AF[kc][t] = (valid && z < 0.0f) ? 1.0f : 0.0f;
      BM[kc][t] = (float)((mbits >> k) & 1u);
    }
  }

  v8f accL = {0.f, 0.f, 0.f, 0.f, 0.f, 0.f, 0.f, 0.f};  // sum log|z| over S_i
  v8f accF = {0.f, 0.f, 0.f, 0.f, 0.f, 0.f, 0.f, 0.f};  // # negative z over S_i
#pragma unroll
  for (int kc = 0; kc < 4; ++kc) {
    accL = __builtin_amdgcn_wmma_f32_16x16x4_f32(
        false, AL[kc], false, BM[kc], (short)0, accL, false, false);
    accF = __builtin_amdgcn_wmma_f32_16x16x4_f32(
        false, AF[kc], false, BM[kc], (short)0, accF, false, false);
  }

  // D layout: VGPR r, lanes 0-15 -> M=r, lanes 16-31 -> M=r+8; N = lane&15.
  if (m < NQ) {
#pragma unroll
    for (int r = 0; r < 8; ++r) {
      const int M = r + 8 * hi;
      const int orow = rowBase + M;
      if (orow < batch) {
        float val = __expf(accL[r]);
        const int negc = (int)(accF[r] + 0.5f);
        val = (negc & 1) ? -val : val;
        out[orow * NQ + m] = val;
      }
    }
  }
}

extern "C" void kernel_launch(void* const* d_in, const int* in_sizes, int n_in,
                              void* d_out, int out_size, void* d_ws, size_t ws_size,
                              hipStream_t stream) {
  const float* x      = (const float*)d_in[0];   // [B, 14] float32
  const float* weight = (const float*)d_in[1];   // [1, 14] float32
  float* out          = (float*)d_out;           // [B, 14] float32
  const int batch = in_sizes[0] / NQ;            // 256
  const int tiles = (batch + 15) / 16;           // 16 waves total
  quantum_layer_wmma_kernel<<<tiles, 32, 0, stream>>>(x, weight, out, batch);
}